// clDiceLoss_16329465659737
// MI455X (gfx1250) — compile-verified
//
#include <hip/hip_runtime.h>
#include <hip/hip_bf16.h>

#define Hh   1024
#define Ww   1024
#define TILE 64
#define HALO 12                 // 11 erodes (L1 radius) + 3x3 dilate => radius 12
#define RS   (TILE + 2*HALO)    // 88
#define RP   92                 // row pitch: 368B = 16B-aligned rows, odd-ish banks
#define NB_RED 1024
#define IMG  (Hh * Ww)

typedef __attribute__((address_space(3))) void lvoid;

// CDNA5 async global->LDS copies (tracked by ASYNCcnt).
__device__ __forceinline__ void async_ld_b32(float* lds_dst, const float* g_src) {
    unsigned lofs = (unsigned)(unsigned long long)(lvoid*)lds_dst;  // AS3 ptr == LDS offset
    asm volatile("global_load_async_to_lds_b32 %0, %1, off"
                 :: "v"(lofs), "v"((unsigned long long)(const void*)g_src)
                 : "memory");
}
__device__ __forceinline__ void async_ld_b128(float* lds_dst, const float* g_src) {
    unsigned lofs = (unsigned)(unsigned long long)(lvoid*)lds_dst;
    asm volatile("global_load_async_to_lds_b128 %0, %1, off"
                 :: "v"(lofs), "v"((unsigned long long)(const void*)g_src)
                 : "memory");
}
__device__ __forceinline__ void wait_async0() {
#if defined(__has_builtin) && __has_builtin(__builtin_amdgcn_s_wait_asynccnt)
    __builtin_amdgcn_s_wait_asynccnt(0);
#else
    asm volatile("s_wait_asynccnt 0" ::: "memory");
#endif
}

// --------------------------------------------------------------------------
// One kernel = entire soft_skeleton for one image set.
// mode 0: src = sigmoid(pred)   mode 1: src = (float)target
// --------------------------------------------------------------------------
__global__ __launch_bounds__(256) void skel_fused(const float* __restrict__ predF,
                                                  const int* __restrict__ tgtI,
                                                  float* __restrict__ S,
                                                  int mode) {
    __shared__ float Ea[RS][RP];
    __shared__ float Eb[RS][RP];

    const int tiles_x = Ww / TILE;              // 16
    const int tpi = tiles_x * (Hh / TILE);      // 256
    int b  = blockIdx.x / tpi;
    int t  = blockIdx.x % tpi;
    int ty = (t / tiles_x) * TILE;
    int tx = (t % tiles_x) * TILE;
    int tid = threadIdx.x;
    size_t base = (size_t)b * IMG;

    bool border = (ty < HALO) || (tx < HALO) ||
                  (ty + TILE + HALO > Hh) || (tx + TILE + HALO > Ww);

    const char* srcB = (mode == 0) ? (const char*)(const void*)predF
                                   : (const char*)(const void*)tgtI;

    // ---- async halo load ----
    if (!border) {
        // 16B-aligned rows: 22 x b128 per 88-float row
        const float* rowbase = (const float*)(const void*)srcB + base
                             + (size_t)(ty - HALO) * Ww + (tx - HALO);
        for (int i = tid; i < RS * 22; i += 256) {
            int ly = i / 22, c = (i % 22) * 4;
            async_ld_b128(&Ea[ly][c], rowbase + (size_t)ly * Ww + c);
        }
    } else {
        for (int i = tid; i < RS * RS; i += 256) {
            int ly = i / RS, lx = i % RS;
            int gy = ty + ly - HALO, gx = tx + lx - HALO;
            int cy = gy < 0 ? 0 : (gy >= Hh ? Hh - 1 : gy);
            int cx = gx < 0 ? 0 : (gx >= Ww ? Ww - 1 : gx);
            const float* g = (const float*)(srcB + 4u * (base + (size_t)cy * Ww + cx));
            async_ld_b32(&Ea[ly][lx], g);
        }
    }
    wait_async0();
    __syncthreads();

    // ---- transform in place: sigmoid / int->float; out-of-image -> +inf ----
    for (int i = tid; i < RS * RS; i += 256) {
        int ly = i / RS, lx = i % RS;
        float r = Ea[ly][lx];
        float v = (mode == 0) ? (1.0f / (1.0f + __expf(-r)))
                              : (float)__float_as_int(r);
        if (border) {
            int gy = ty + ly - HALO, gx = tx + lx - HALO;
            if (gy < 0 || gy >= Hh || gx < 0 || gx >= Ww)
                v = __builtin_inff();
        }
        Ea[ly][lx] = v;
    }
    __syncthreads();

    float (*cur)[RP] = Ea;      // E_k
    float (*nxt)[RP] = Eb;      // E_{k+1}

    float sk[16];
#pragma unroll
    for (int r = 0; r < 16; ++r) sk[r] = 0.0f;

    int px  = tid & 63;         // 0..63
    int py0 = tid >> 6;         // 0..3

    for (int k = 0; k < 11; ++k) {
        // ---- erode cur -> nxt; rows shrink with k: y in [k+1, 87-k) ----
        int rows = 86 - 2 * k;
        for (int i = tid; i < rows * 86; i += 256) {
            int y = i / 86 + (k + 1), x = i % 86 + 1;
            float e = fminf(fminf(fminf(cur[y - 1][x], cur[y + 1][x]),
                                  fminf(cur[y][x - 1], cur[y][x + 1])),
                            cur[y][x]);
            if (border) {
                int gy = ty + y - HALO, gx = tx + x - HALO;
                if (gy < 0 || gy >= Hh || gx < 0 || gx >= Ww)
                    e = __builtin_inff();      // keep out-of-image marked
            }
            nxt[y][x] = e;
        }
        __syncthreads();

        // ---- dilate(E_{k+1}) at tile pixels + skeleton update (registers) ----
#pragma unroll
        for (int r = 0; r < 16; ++r) {
            int cy = (py0 + 4 * r) + HALO, cx = px + HALO;
            float m;
            if (border) {
                m = -1.0f;                      // all real values are in [0,1]
#pragma unroll
                for (int dy = -1; dy <= 1; ++dy)
#pragma unroll
                    for (int dx = -1; dx <= 1; ++dx) {
                        float q = nxt[cy + dy][cx + dx];
                        q = (q > 2.0f) ? -1.0f : q;   // +inf marker -> -inf pad
                        m = fmaxf(m, q);
                    }
            } else {
                m = fmaxf(fmaxf(nxt[cy - 1][cx - 1], nxt[cy - 1][cx]), nxt[cy - 1][cx + 1]);
                m = fmaxf(m, fmaxf(fmaxf(nxt[cy][cx - 1], nxt[cy][cx]), nxt[cy][cx + 1]));
                m = fmaxf(m, fmaxf(fmaxf(nxt[cy + 1][cx - 1], nxt[cy + 1][cx]), nxt[cy + 1][cx + 1]));
            }
            float xc    = cur[cy][cx];          // E_k at pixel (finite)
            float delta = fmaxf(xc - m, 0.0f);
            float s     = sk[r];
            sk[r] = s + fmaxf(delta - s * delta, 0.0f);
        }
        __syncthreads();

        float (*tmp)[RP] = cur; cur = nxt; nxt = tmp;
    }

    // ---- store final skeleton (write-only) ----
#pragma unroll
    for (int r = 0; r < 16; ++r) {
        int py = py0 + 4 * r;
        S[base + (size_t)(ty + py) * Ww + (tx + px)] = sk[r];
    }
}

// ----------------------------------------------------------- reductions -----
__device__ __forceinline__ float block_reduce(float v, float* red) {
    int tid = threadIdx.x;
    red[tid] = v;
    __syncthreads();
    for (int off = 128; off > 0; off >>= 1) {
        if (tid < off) red[tid] += red[tid + off];
        __syncthreads();
    }
    float r = red[0];
    __syncthreads();
    return r;
}

__global__ __launch_bounds__(256) void reduce7(const float* __restrict__ pred,
                                               const int* __restrict__ tgt,
                                               const float* __restrict__ SP,
                                               const float* __restrict__ ST,
                                               float* __restrict__ partials,
                                               size_t n) {
    __shared__ float red[256];
    float s0 = 0, s1 = 0, s2 = 0, s3 = 0, s4 = 0, s5 = 0, s6 = 0;
    size_t stride = (size_t)gridDim.x * blockDim.x;
    for (size_t i = (size_t)blockIdx.x * blockDim.x + threadIdx.x; i < n; i += stride) {
        float p  = 1.0f / (1.0f + __expf(-pred[i]));
        float t  = (float)tgt[i];
        float sp = SP[i], st = ST[i];
        s0 += p;  s1 += t;  s2 += p * t;
        s3 += sp; s4 += sp * t;
        s5 += st; s6 += st * p;
    }
    float sums[7] = {s0, s1, s2, s3, s4, s5, s6};
    for (int q = 0; q < 7; ++q) {
        float r = block_reduce(sums[q], red);
        if (threadIdx.x == 0) partials[(size_t)q * gridDim.x + blockIdx.x] = r;
    }
}

__global__ __launch_bounds__(256) void finalize(const float* __restrict__ partials,
                                                float* __restrict__ out) {
    __shared__ float red[256];
    float tot[7];
    for (int q = 0; q < 7; ++q) {
        float local = 0.0f;
        for (int i = threadIdx.x; i < NB_RED; i += 256)
            local += partials[(size_t)q * NB_RED + i];
        tot[q] = block_reduce(local, red);
    }
    if (threadIdx.x == 0) {
        float sumP = tot[0], sumT = tot[1], sumPT = tot[2];
        float sumSP = tot[3], sumSPT = tot[4];
        float sumST = tot[5], sumSTP = tot[6];
        float tprec = (sumSPT + 1.0f) / (sumSP + 1.0f);
        float tsens = (sumSTP + 1.0f) / (sumST + 1.0f);
        float cl    = 2.0f * tprec * tsens / (tprec + tsens + 1e-7f);
        float dice  = (2.0f * sumPT + 1.0f) / (sumP + sumT + 1.0f);
        out[0] = 1.0f - (0.5f * dice + 0.5f * cl);
    }
}

// -------------------------------------------------------------- launcher ----
extern "C" void kernel_launch(void* const* d_in, const int* in_sizes, int n_in,
                              void* d_out, int out_size, void* d_ws, size_t ws_size,
                              hipStream_t stream) {
    (void)n_in; (void)out_size; (void)ws_size;
    const float* pred = (const float*)d_in[0];
    const int*   tgt  = (const int*)d_in[1];
    size_t n  = (size_t)in_sizes[0];
    int nimg  = (int)(n / (size_t)IMG);

    float* ws = (float*)d_ws;
    float* SP = ws;
    float* ST = SP + n;
    float* partials = ST + n;        // 7 * NB_RED floats

    int nblk = nimg * (Hh / TILE) * (Ww / TILE);   // 8 * 256 = 2048

    skel_fused<<<nblk, 256, 0, stream>>>(pred, tgt, SP, 0);
    skel_fused<<<nblk, 256, 0, stream>>>(pred, tgt, ST, 1);

    reduce7<<<NB_RED, 256, 0, stream>>>(pred, tgt, SP, ST, partials, n);
    finalize<<<1, 256, 0, stream>>>(partials, (float*)d_out);
}